// TinyGRUVAD_73134703116826
// MI455X (gfx1250) — compile-verified
//
#include <hip/hip_runtime.h>
#include <hip/hip_bf16.h>

typedef __attribute__((ext_vector_type(2))) float v2f;
typedef __attribute__((ext_vector_type(4))) float v4f;
typedef __attribute__((ext_vector_type(8))) float v8f;

#define GRU_B 256
#define GRU_T 4096
#define GRU_C 32
#define GRU_H 16
#define LN_EPS 1e-5f

static __device__ __forceinline__ v8f wmma_f32_k4(v2f a, v2f b, v8f c) {
    return __builtin_amdgcn_wmma_f32_16x16x4_f32(false, a, false, b, (short)0, c,
                                                 false, false);
}

// ---------------------------------------------------------------------------
// Kernel 1: conv1d(k=3, causal) + LayerNorm(C=32) + gx = yn @ w_ih^T + b_ih
// One wave per (batch, 16-timestep tile). Writes gx to ws as [t_local][b][48].
// Memory-bound, full-device parallelism.
// ---------------------------------------------------------------------------
__global__ __launch_bounds__(32)
void TinyGRUVAD_gx_kernel(const float* __restrict__ x,
                          const float* __restrict__ conv_w,   // [C,3]
                          const float* __restrict__ conv_b,   // [C]
                          const float* __restrict__ ln_g,     // [C]
                          const float* __restrict__ ln_b,     // [C]
                          const float* __restrict__ w_ih,     // [48,32]
                          const float* __restrict__ b_ih,     // [48]
                          float* __restrict__ gxws,           // [chunk][256][48]
                          int t0) {
    const int lane = threadIdx.x;
    const int tt   = blockIdx.x;          // 16-t tile within chunk
    const int b    = blockIdx.y;
    const int tbase = t0 + tt * 16;       // global t of tile row 0

    __shared__ float xt[18 * GRU_C];      // rows tbase-2 .. tbase+15
    __shared__ float ynt[16 * GRU_C];

    // --- coalesced load of 18 input rows (zero-pad for global t < 0) ---
    const float* xb = x + (size_t)b * GRU_T * GRU_C;
    for (int i = lane; i < 18 * GRU_C; i += 32) {
        int gr = tbase - 2 + (i >> 5);    // global row
        xt[i] = (gr >= 0) ? xb[(size_t)gr * GRU_C + (i & 31)] : 0.0f;
    }
    __syncthreads();

    // --- conv + LN: lane pair (l, l+16) handles row t = l%16, 16 chans each ---
    const int tr  = lane & 15;
    const int hi  = lane >> 4;            // 0: chans 0..15, 1: chans 16..31
    const int ch0 = hi * 16;
    float y[16];
    float s = 0.0f, s2 = 0.0f;
    for (int i = 0; i < 16; ++i) {
        int c = ch0 + i;
        float v = xt[tr * GRU_C + c]        * conv_w[c * 3 + 0]
                + xt[(tr + 1) * GRU_C + c]  * conv_w[c * 3 + 1]
                + xt[(tr + 2) * GRU_C + c]  * conv_w[c * 3 + 2]
                + conv_b[c];
        y[i] = v; s += v; s2 += v * v;
    }
    s  += __shfl_xor(s, 16, 32);
    s2 += __shfl_xor(s2, 16, 32);
    float mu   = s * (1.0f / 32.0f);
    float var  = s2 * (1.0f / 32.0f) - mu * mu;
    float rstd = rsqrtf(var + LN_EPS);
    for (int i = 0; i < 16; ++i) {
        int c = ch0 + i;
        ynt[tr * GRU_C + c] = (y[i] - mu) * rstd * ln_g[c] + ln_b[c];
    }
    __syncthreads();

    // --- A slices: Yn [16t x 32c], f32 WMMA 16x16x4 layout ---
    v2f A[8];
    for (int ks = 0; ks < 8; ++ks) {
        int c0 = 4 * ks + 2 * hi;
        A[ks] = *(const v2f*)&ynt[tr * GRU_C + c0];
    }

    // --- per gate: 8 chained K=4 WMMAs, then bias + store in D layout ---
    for (int g = 0; g < 3; ++g) {
        v8f acc = {};
        for (int ks = 0; ks < 8; ++ks) {
            int c0 = 4 * ks + 2 * hi;
            v2f Bv = *(const v2f*)&w_ih[(g * 16 + tr) * GRU_C + c0];
            acc = wmma_f32_k4(A[ks], Bv, acc);
        }
        float bias = b_ih[g * 16 + tr];
        for (int v = 0; v < 8; ++v) {
            int tl = tt * 16 + v + 8 * hi;                    // local t in chunk
            gxws[((size_t)tl * GRU_B + b) * 48 + g * 16 + tr] = acc[v] + bias;
        }
    }
}

// ---------------------------------------------------------------------------
// Kernel 2: sequential GRU scan over one T-chunk + fc head.
// One wave per 16 batch elements; h lives in registers in WMMA C/D layout
// (v8f: element v on lane l -> h[j = v + 8*(l>=16)][batch = l%16]).
// Each step: 3 gates x 4 chained v_wmma_f32_16x16x4_f32 for gh = h @ w_hh^T.
// gx loads are software-pipelined one step ahead (off the dependency chain).
// ---------------------------------------------------------------------------
__global__ __launch_bounds__(32)
void TinyGRUVAD_scan_kernel(const float* __restrict__ gxws,  // [chunk][256][48]
                            const float* __restrict__ h0,    // [1,B,H]
                            const float* __restrict__ w_hh,  // [48,16]
                            const float* __restrict__ b_hh,  // [48]
                            const float* __restrict__ fc_w,  // [1,16]
                            const float* __restrict__ fc_b,  // [1]
                            float* __restrict__ out,         // logits + h_last
                            float* __restrict__ hstate,      // [16][32][8]
                            int t0, int chunk_t, int first, int last) {
    const int lane = threadIdx.x;
    const int bg   = blockIdx.x;
    const int tr   = lane & 15;           // batch within group (N index)
    const int hi   = lane >> 4;
    const int b    = bg * 16 + tr;
    const bool lo  = (hi == 0);

    // w_hh A-slices: A_g[ks] = [16j x 4k] slice, resident for the whole scan
    v2f A[3][4];
    for (int g = 0; g < 3; ++g)
        for (int ks = 0; ks < 4; ++ks)
            A[g][ks] = *(const v2f*)&w_hh[(g * 16 + tr) * GRU_H + 4 * ks + 2 * hi];

    float bhh[3][8];
    for (int g = 0; g < 3; ++g)
        for (int v = 0; v < 8; ++v)
            bhh[g][v] = b_hh[g * 16 + 8 * hi + v];
    float fcw[8];
    for (int v = 0; v < 8; ++v) fcw[v] = fc_w[8 * hi + v];
    const float fcb = fc_b[0];

    // hidden state in D layout
    v8f h;
    if (first) {
        const float* hp = h0 + (size_t)b * GRU_H + 8 * hi;
        for (int v = 0; v < 8; ++v) h[v] = hp[v];
    } else {
        const float* hs = hstate + ((size_t)bg * 32 + lane) * 8;
        for (int v = 0; v < 8; ++v) h[v] = hs[v];
    }

    const size_t lane_off = (size_t)b * 48 + 8 * hi;

    // prime the gx pipeline with step 0
    v4f r0, r1, z0, z1, n0, n1;
    {
        const v4f* p = (const v4f*)(gxws + lane_off);
        r0 = p[0]; r1 = p[1]; z0 = p[4]; z1 = p[5]; n0 = p[8]; n1 = p[9];
    }

    for (int tl = 0; tl < chunk_t; ++tl) {
        // current step's gx (loaded one iteration ago)
        v8f gx[3];
        for (int v = 0; v < 4; ++v) {
            gx[0][v] = r0[v]; gx[0][v + 4] = r1[v];
            gx[1][v] = z0[v]; gx[1][v + 4] = z1[v];
            gx[2][v] = n0[v]; gx[2][v + 4] = n1[v];
        }

        // issue next step's gx loads now (uniform clamped index, no divergence)
        {
            int tn = (tl + 1 < chunk_t) ? (tl + 1) : tl;
            const v4f* p = (const v4f*)(gxws + (size_t)tn * (GRU_B * 48) + lane_off);
            r0 = p[0]; r1 = p[1]; z0 = p[4]; z1 = p[5]; n0 = p[8]; n1 = p[9];
            // pull a later step from L2 toward the WGP ahead of time
            int tp = (tl + 8 < chunk_t) ? (tl + 8) : tl;
            __builtin_prefetch(gxws + (size_t)tp * (GRU_B * 48) + lane_off, 0, 3);
        }

        // reshape h (D layout) -> B-operand K-slices via half-wave swap
        v8f hx;
        for (int v = 0; v < 8; ++v) hx[v] = __shfl_xor(h[v], 16, 32);
        v2f B0 = { lo ? h[0]  : hx[2], lo ? h[1]  : hx[3] };   // k = 0..3
        v2f B1 = { lo ? h[4]  : hx[6], lo ? h[5]  : hx[7] };   // k = 4..7
        v2f B2 = { lo ? hx[0] : h[2],  lo ? hx[1] : h[3]  };   // k = 8..11
        v2f B3 = { lo ? hx[4] : h[6],  lo ? hx[5] : h[7]  };   // k = 12..15

        // gh_g = W_g (16x16) @ H (16x16): 4 chained K=4 f32 WMMAs per gate
        v8f gh[3];
        for (int g = 0; g < 3; ++g) {
            v8f acc = {};
            acc = wmma_f32_k4(A[g][0], B0, acc);
            acc = wmma_f32_k4(A[g][1], B1, acc);
            acc = wmma_f32_k4(A[g][2], B2, acc);
            acc = wmma_f32_k4(A[g][3], B3, acc);
            gh[g] = acc;
        }

        // elementwise GRU update + fc partial dot
        v8f hn;
        float pdot = 0.0f;
        for (int v = 0; v < 8; ++v) {
            float r  = 1.0f / (1.0f + __expf(-(gx[0][v] + gh[0][v] + bhh[0][v])));
            float z  = 1.0f / (1.0f + __expf(-(gx[1][v] + gh[1][v] + bhh[1][v])));
            float na = gx[2][v] + r * (gh[2][v] + bhh[2][v]);
            float nt = 2.0f / (1.0f + __expf(-2.0f * na)) - 1.0f;   // tanh
            float hv = (1.0f - z) * nt + z * h[v];
            hn[v] = hv;
            pdot += fcw[v] * hv;
        }
        h = hn;
        pdot += __shfl_xor(pdot, 16, 32);
        if (lane < 16) out[(size_t)b * GRU_T + (t0 + tl)] = pdot + fcb;
    }

    // persist h across chunks
    float* hs = hstate + ((size_t)bg * 32 + lane) * 8;
    for (int v = 0; v < 8; ++v) hs[v] = h[v];
    if (last) {
        float* o = out + (size_t)GRU_B * GRU_T + (size_t)b * GRU_H + 8 * hi;
        for (int v = 0; v < 8; ++v) o[v] = h[v];
    }
}

extern "C" void kernel_launch(void* const* d_in, const int* in_sizes, int n_in,
                              void* d_out, int out_size, void* d_ws, size_t ws_size,
                              hipStream_t stream) {
    const float* x      = (const float*)d_in[0];
    const float* h0     = (const float*)d_in[1];
    const float* conv_w = (const float*)d_in[2];
    const float* conv_b = (const float*)d_in[3];
    const float* ln_g   = (const float*)d_in[4];
    const float* ln_b   = (const float*)d_in[5];
    const float* w_ih   = (const float*)d_in[6];
    const float* w_hh   = (const float*)d_in[7];
    const float* b_ih   = (const float*)d_in[8];
    const float* b_hh   = (const float*)d_in[9];
    const float* fc_w   = (const float*)d_in[10];
    const float* fc_b   = (const float*)d_in[11];
    float* out = (float*)d_out;

    const size_t per_t = (size_t)GRU_B * 48 * sizeof(float);
    int chunk = 512;
    while (chunk > 32 &&
           (size_t)chunk * per_t + (size_t)GRU_B * GRU_H * sizeof(float) > ws_size)
        chunk >>= 1;

    float* gxws   = (float*)d_ws;
    float* hstate = (float*)((char*)d_ws + (size_t)chunk * per_t);

    for (int t0 = 0; t0 < GRU_T; t0 += chunk) {
        dim3 g1(chunk / 16, GRU_B);
        TinyGRUVAD_gx_kernel<<<g1, 32, 0, stream>>>(
            x, conv_w, conv_b, ln_g, ln_b, w_ih, b_ih, gxws, t0);
        TinyGRUVAD_scan_kernel<<<GRU_B / 16, 32, 0, stream>>>(
            gxws, h0, w_hh, b_hh, fc_w, fc_b, out, hstate,
            t0, chunk, (t0 == 0) ? 1 : 0, (t0 + chunk >= GRU_T) ? 1 : 0);
    }
}